// EncoderModelLoss_1589137899697
// MI455X (gfx1250) — compile-verified
//
#include <hip/hip_runtime.h>
#include <hip/hip_bf16.h>
#include <math.h>

// ---------------------------------------------------------------------------
// Binary cross-entropy loss + accuracy over E = 16,777,216 rows of (2 logits,
// 1 label). Pure streaming reduction: ~200 MB read, ~9 us at 23.3 TB/s HBM.
// Memory-bound -> 128-bit coalesced loads + f32 accumulation.
// The 32-lane wave reduction uses V_WMMA_F32_16X16X4_F32 (f32 in/out, exact):
//   A (16x4 f32, 2 VGPRs/lane): VGPR0 = lane's nll partial, VGPR1 = acc partial
//   B (4x16 f32, 2 VGPRs):  rows 0,2 = indicator(col<8), rows 1,3 = indicator(col>=8)
//   => D[m][n<8]  = nll[m] + nll[m+16],  D[m][n>=8] = acc[m] + acc[m+16]
// Summing the 8 D VGPRs + one shfl_xor(16) yields: lanes 0-7 = wave nll sum,
// lanes 8-15 = wave acc sum. Deterministic, numerically exact f32 adds.
// ---------------------------------------------------------------------------

typedef __attribute__((ext_vector_type(2))) float v2f;
typedef __attribute__((ext_vector_type(8))) float v8f;

#if defined(__has_builtin)
#if __has_builtin(__builtin_amdgcn_wmma_f32_16x16x4_f32)
#define HAVE_WMMA_F32X4 1
#endif
#endif

#define NBLOCKS   2048
#define NTHREADS  256
#define WAVES_PER_BLOCK (NTHREADS / 32)

__device__ __forceinline__ void row_update(float r0, float r1, int lab,
                                           float& nll, float& acc) {
    float m   = fmaxf(r0, r1);
    float lse = m + __logf(__expf(r0 - m) + __expf(r1 - m));
    nll += lse - (lab ? r1 : r0);
    bool correct = (lab == 0) ? (r0 > r1) : (r0 < r1);
    acc += correct ? 1.0f : 0.0f;
}

// Reduce (nll, acc) across the 32 lanes of a wave. On return, lane 0 holds
// both wave sums.
__device__ __forceinline__ void wave_reduce(float& nll, float& acc) {
#if defined(HAVE_WMMA_F32X4)
    unsigned lane = threadIdx.x & 31u;
    v2f A;
    A.x = nll;                                   // A[m][0] / A[m][2]
    A.y = acc;                                   // A[m][1] / A[m][3]
    v2f B;
    float lo = ((lane & 15u) < 8u) ? 1.0f : 0.0f;
    B.x = lo;                                    // B rows 0,2: select nll -> cols 0-7
    B.y = 1.0f - lo;                             // B rows 1,3: select acc -> cols 8-15
    v8f C = {};
    v8f D = __builtin_amdgcn_wmma_f32_16x16x4_f32(
        /*neg_a=*/false, A, /*neg_b=*/false, B,
        /*c_mod=*/(short)0, C, /*reuse_a=*/false, /*reuse_b=*/false);
    float t = ((D[0] + D[1]) + (D[2] + D[3])) + ((D[4] + D[5]) + (D[6] + D[7]));
    float s = t + __shfl_xor(t, 16, 32);         // fold upper/lower row halves
    // lanes 0-7: full wave nll sum; lanes 8-15: full wave acc sum
    nll = __shfl(s, 0, 32);
    acc = __shfl(s, 8, 32);
#else
    for (int off = 16; off > 0; off >>= 1) {
        nll += __shfl_xor(nll, off, 32);
        acc += __shfl_xor(acc, off, 32);
    }
#endif
}

__global__ __launch_bounds__(NTHREADS) void ce_partial_kernel(
    const float* __restrict__ result,   // [n_rows, 2] f32
    const int*   __restrict__ labels,   // [n_rows] int
    float*       __restrict__ partials, // [gridDim.x, 2] f32
    unsigned n_rows)
{
    const unsigned tid    = blockIdx.x * blockDim.x + threadIdx.x;
    const unsigned stride = gridDim.x * blockDim.x;

    const float4* __restrict__ res4 = reinterpret_cast<const float4*>(result);
    const int4*   __restrict__ lab4 = reinterpret_cast<const int4*>(labels);

    float nll = 0.0f, acc = 0.0f;

    // 4 rows per iteration: two b128 logit loads + one b128 label load,
    // consecutive threads touch consecutive 32B/16B chunks (fully coalesced).
    const unsigned n_quads = n_rows >> 2;
    for (unsigned q = tid; q < n_quads; q += stride) {
        float4 ab = res4[2u * q];        // rows 4q, 4q+1
        float4 cd = res4[2u * q + 1u];   // rows 4q+2, 4q+3
        int4   lb = lab4[q];
        row_update(ab.x, ab.y, lb.x, nll, acc);
        row_update(ab.z, ab.w, lb.y, nll, acc);
        row_update(cd.x, cd.y, lb.z, nll, acc);
        row_update(cd.z, cd.w, lb.w, nll, acc);
    }

    // Intra-wave reduction via WMMA (EXEC is all ones here: no divergence).
    wave_reduce(nll, acc);

    // Cross-wave (block) reduction through LDS.
    __shared__ float s_nll[WAVES_PER_BLOCK];
    __shared__ float s_acc[WAVES_PER_BLOCK];
    const unsigned lane = threadIdx.x & 31u;
    const unsigned wave = threadIdx.x >> 5;
    if (lane == 0) { s_nll[wave] = nll; s_acc[wave] = acc; }
    __syncthreads();
    if (threadIdx.x == 0) {
        float bn = 0.0f, ba = 0.0f;
        #pragma unroll
        for (int w = 0; w < WAVES_PER_BLOCK; ++w) { bn += s_nll[w]; ba += s_acc[w]; }
        partials[2u * blockIdx.x]      = bn;
        partials[2u * blockIdx.x + 1u] = ba;
    }
}

__global__ __launch_bounds__(NTHREADS) void ce_finalize_kernel(
    const float* __restrict__ partials, int n_partials,
    float* __restrict__ out, float inv_n)
{
    __shared__ float sn[NTHREADS];
    __shared__ float sa[NTHREADS];
    float n0 = 0.0f, a0 = 0.0f;
    for (int i = threadIdx.x; i < n_partials; i += NTHREADS) {
        n0 += partials[2 * i];
        a0 += partials[2 * i + 1];
    }
    sn[threadIdx.x] = n0;
    sa[threadIdx.x] = a0;
    __syncthreads();
    for (int s = NTHREADS / 2; s > 0; s >>= 1) {
        if ((int)threadIdx.x < s) {
            sn[threadIdx.x] += sn[threadIdx.x + s];
            sa[threadIdx.x] += sa[threadIdx.x + s];
        }
        __syncthreads();
    }
    if (threadIdx.x == 0) {
        out[0] = sn[0] * inv_n;   // loss
        out[1] = sa[0] * inv_n;   // accuracy
    }
}

extern "C" void kernel_launch(void* const* d_in, const int* in_sizes, int n_in,
                              void* d_out, int out_size, void* d_ws, size_t ws_size,
                              hipStream_t stream) {
    const float* result = (const float*)d_in[0];   // [E,2] f32
    const int*   labels = (const int*)d_in[1];     // [E] int
    float* out = (float*)d_out;                    // {loss, accuracy}
    float* partials = (float*)d_ws;

    const unsigned n_rows = (unsigned)in_sizes[1]; // E = 16,777,216

    // Stay inside workspace if it is smaller than expected.
    int nblocks = NBLOCKS;
    size_t need = (size_t)nblocks * 2u * sizeof(float);
    if (need > ws_size && ws_size >= 2u * sizeof(float)) {
        nblocks = (int)(ws_size / (2u * sizeof(float)));
        if (nblocks > NBLOCKS) nblocks = NBLOCKS;
        if (nblocks < 1) nblocks = 1;
    }

    ce_partial_kernel<<<nblocks, NTHREADS, 0, stream>>>(result, labels, partials, n_rows);
    ce_finalize_kernel<<<1, NTHREADS, 0, stream>>>(partials, nblocks, out, 1.0f / (float)n_rows);
}